// SoundWave2DModel_16887811408546
// MI455X (gfx1250) — compile-verified
//
#include <hip/hip_runtime.h>

#define NZ 512
#define NX 512
#define DT2 0.0025f   // DT*DT, DT=0.05; DZ=1 so no further scaling

typedef __attribute__((ext_vector_type(2))) float v2f;
typedef __attribute__((ext_vector_type(8))) float v8f;

// ---------------------------------------------------------------------------
// Prep: pack all step-invariant per-cell coefficients into one float4 field:
//   C4 = { A1 = dt2*Kc*rho,  A2x = dt2*Kc*gxr,  A2z = dt2*Kc*gzr,  dcoef = 2*beta/Kc }
// with gxr/gzr = central grads of 1/rho (zero-padded), Kc = v^2*rho.
// ---------------------------------------------------------------------------
__global__ void prep_kernel(const float* __restrict__ vel,
                            const float* __restrict__ rho,
                            const float* __restrict__ beta,
                            float4* __restrict__ C4) {
    int i = blockIdx.x * blockDim.x + threadIdx.x;
    if (i >= NZ * NX) return;
    int z = i / NX, x = i - z * NX;
    auto ir = [&](int zz, int xx) -> float {
        return (zz >= 0 && zz < NZ && xx >= 0 && xx < NX) ? 1.0f / rho[zz * NX + xx] : 0.0f;
    };
    float Kc  = vel[i] * vel[i] * rho[i];
    float gxr = 0.5f * (ir(z, x + 1) - ir(z, x - 1));
    float gzr = 0.5f * (ir(z + 1, x) - ir(z - 1, x));
    float4 c;
    c.x = DT2 * Kc * rho[i];
    c.y = DT2 * Kc * gxr;
    c.z = DT2 * Kc * gzr;
    c.w = 2.0f * beta[i] / Kc;
    C4[i] = c;
}

// ---------------------------------------------------------------------------
// One time step. Each wave owns a 16x16 tile; 8 waves/block -> 128 blocks.
// Laplacian + gradients as banded matmuls via v_wmma_f32_16x16x4_f32.
// Block 0 additionally gathers the receiver traces of the PREVIOUS step
// (pm1 == p_{t-1}, complete by launch ordering) -> no separate gather nodes.
// ---------------------------------------------------------------------------
__launch_bounds__(256)
__global__ void step_kernel(const float* __restrict__ pm1, const float* __restrict__ pm2,
                            float* __restrict__ pout,
                            const float4* __restrict__ C4,
                            const float* __restrict__ damping,
                            const float* __restrict__ pulse_t,
                            const float* __restrict__ pulse_next,
                            const int* __restrict__ nlaflag,
                            const int* __restrict__ el,
                            float* __restrict__ gout,  // nullptr or d_out row for p_{t-1}
                            int nch) {
    __shared__ float T[8][20 * 21];   // per-wave 20x20 halo tile, stride 21
    __shared__ float TW[40];          // (d+18) -> laplacian weight, zero-padded
    __shared__ float TG[40];          // (d+18) -> gradient  weight, zero-padded

    const int wave = threadIdx.x >> 5;
    const int lane = threadIdx.x & 31;
    const int tile = blockIdx.x * 8 + wave;   // 32x32 tiles of 16x16
    const int Z0 = (tile >> 5) << 4;
    const int X0 = (tile & 31) << 4;

    // Free-ride receiver gather for the previous step's field.
    if (gout != nullptr && blockIdx.x == 0 && threadIdx.x < nch) {
        int iz = el[2 * threadIdx.x + 0];
        int ix = el[2 * threadIdx.x + 1];
        gout[threadIdx.x] = pm1[iz * NX + ix];
    }

    // Build the banded-weight lookup tables once per block.
    for (int i = threadIdx.x; i < 40; i += 256) {
        int d = i - 18;
        float w = 0.0f, g = 0.0f;
        if (d == -2 || d == 2) w = -1.0f / 12.0f;
        else if (d == -1 || d == 1) w = 4.0f / 3.0f;
        else if (d == 0) w = -2.5f;
        if (d == 1) g = 0.5f;
        else if (d == -1) g = -0.5f;
        TW[i] = w;
        TG[i] = g;
    }

    // Prefetch next step's pulse rows for this tile (the only HBM-resident
    // stream; everything else is L2-resident). One lane per tile row.
    if (lane < 16) {
        __builtin_prefetch(&pulse_next[(Z0 + lane) * NX + X0], 0, 3);
    }

    // Stage pm1 halo tile (zero-padded boundary) into LDS.
    for (int i = lane; i < 400; i += 32) {
        int zz = i / 20, xx = i - zz * 20;
        int gz = Z0 - 2 + zz, gx = X0 - 2 + xx;
        float v = (gz >= 0 && gz < NZ && gx >= 0 && gx < NX) ? pm1[gz * NX + gx] : 0.0f;
        T[wave][zz * 21 + xx] = v;
    }
    __syncthreads();

    const int half = lane >> 4;            // 0..1
    const int lm   = lane & 15;            // 0..15
    const int wbase = 2 * half - lm + 16;  // (d+18) for c=0,r=0; stays within [0,40)

    v8f accL  = {};  // Lx + Lz  (full 4th-order Laplacian)
    v8f accGx = {};  // grad_x(pm1)
    v8f accGz = {};  // grad_z(pm1)

#pragma unroll
    for (int c = 0; c < 5; ++c) {
        const int k0 = 4 * c + 2 * half;   // K index base for this lane/fragment
        const int wi = 4 * c + wbase;

        // Weight fragments straight from split tables: one 2-address LDS load
        // each, no register transposes. Same content serves A (z) and B (x).
        v2f fw; fw[0] = TW[wi]; fw[1] = TW[wi + 1];
        v2f fg; fg[0] = TG[wi]; fg[1] = TG[wi + 1];

        // P fragments from LDS tile: Px (A-layout, rows=z), Pz (B-layout, cols=x)
        v2f px, pz;
        px[0] = T[wave][(lm + 2) * 21 + k0];
        px[1] = T[wave][(lm + 2) * 21 + k0 + 1];
        pz[0] = T[wave][k0 * 21 + lm + 2];
        pz[1] = T[wave][(k0 + 1) * 21 + lm + 2];

        accL  = __builtin_amdgcn_wmma_f32_16x16x4_f32(false, px, false, fw, (short)0, accL,  false, false); // Lx
        accL  = __builtin_amdgcn_wmma_f32_16x16x4_f32(false, fw, false, pz, (short)0, accL,  false, false); // Lz
        accGx = __builtin_amdgcn_wmma_f32_16x16x4_f32(false, px, false, fg, (short)0, accGx, false, false);
        accGz = __builtin_amdgcn_wmma_f32_16x16x4_f32(false, fg, false, pz, (short)0, accGz, false, false);
    }

    const int NLA = nlaflag[0];

#pragma unroll
    for (int r = 0; r < 8; ++r) {
        int m   = r + 8 * half;            // C/D layout: M = vgpr + 8*(lane/16)
        int gz  = Z0 + m, gx = X0 + lm;
        int idx = gz * NX + gx;

        float  p1  = T[wave][(m + 2) * 21 + lm + 2];
        float4 cf  = C4[idx];              // {A1, A2x, A2z, dcoef} : one b128 load
        float  dmp = damping[idx];
        float  p2  = pm2[idx];
        float  pu  = pulse_t[idx];

        float rhs = cf.x * accL[r] + cf.y * accGx[r] + cf.z * accGz[r];

        float pnew;
        if (NLA) {
            float nla = 1.0f + cf.w * p1;
            float a   = nla + 2.0f * dmp;
            float b   = 2.0f * nla - dmp * dmp;
            float ccc = 2.0f * dmp - nla;
            float dp  = p1 - p2;
            pnew = (b * p1 + ccc * p2 + cf.w * dp * dp + rhs + pu) / a;
        } else {
            float a   = 1.0f + 2.0f * dmp;
            float b   = 2.0f - dmp * dmp;
            float ccc = 2.0f * dmp - 1.0f;
            pnew = (b * p1 + ccc * p2 + rhs + DT2 * pu) / a;
        }
        pout[idx] = pnew;
    }
}

extern "C" void kernel_launch(void* const* d_in, const int* in_sizes, int n_in,
                              void* d_out, int out_size, void* d_ws, size_t ws_size,
                              hipStream_t stream) {
    const float* pulse   = (const float*)d_in[0];
    const float* vel     = (const float*)d_in[1];
    const float* rho     = (const float*)d_in[2];
    const float* damping = (const float*)d_in[3];
    const float* beta    = (const float*)d_in[4];
    const int*   el      = (const int*)d_in[5];
    const int*   nla     = (const int*)d_in[6];

    const int nt  = in_sizes[0] / (NZ * NX);   // 128
    const int nch = in_sizes[5] / 2;           // 128

    float* ws = (float*)d_ws;
    const size_t F = (size_t)NZ * NX;
    float*  bufs[3] = { ws, ws + F, ws + 2 * F };
    float4* C4 = (float4*)(ws + 3 * F);        // occupies 4F floats

    // Zero the two initial wave fields (B0,B1 contiguous). Capture-safe memset node.
    hipMemsetAsync(bufs[0], 0, 2 * F * sizeof(float), stream);

    prep_kernel<<<(int)((F + 255) / 256), 256, 0, stream>>>(vel, rho, beta, C4);

    // Rotation: pm2_{t+1} = pm1_t, pm1_{t+1} = p_t, write into the retiring buffer.
    // Sample rows: p_t for t = 0,4,...  Each is gathered one launch later from pm1.
    int i1 = 0, i2 = 1;
    for (int t = 0; t < nt; ++t) {
        int io = 3 - i1 - i2;
        const float* pu_t = pulse + (size_t)t * F;
        const float* pu_n = (t + 1 < nt) ? pulse + (size_t)(t + 1) * F : pu_t;
        float* gout = (t >= 1 && ((t - 1) & 3) == 0)
                    ? (float*)d_out + (size_t)((t - 1) >> 2) * nch : nullptr;
        step_kernel<<<128, 256, 0, stream>>>(bufs[i1], bufs[i2], bufs[io],
                                             C4, damping, pu_t, pu_n, nla,
                                             el, gout, nch);
        i2 = i1; i1 = io;
    }
}